// SPINN_12266426597536
// MI455X (gfx1250) — compile-verified
//
#include <hip/hip_runtime.h>
#include <hip/hip_bf16.h>
#include <math.h>

typedef float v2f __attribute__((ext_vector_type(2)));
typedef float v8f __attribute__((ext_vector_type(8)));

// Sizes fixed by the reference: N=256, D=3, H=256, M=2, R=128, M*R=256.

// ---------------------------------------------------------------------------
// CDNA5 async copy helpers (ASYNCcnt-tracked global->LDS DMA, ISA ch.10/15.18)
// LDS dest address = low 32 bits of the generic pointer (aperture layout puts
// the workgroup-relative LDS offset in addr[31:0]).
// ---------------------------------------------------------------------------
__device__ __forceinline__ unsigned lds_off(const void* p) {
  return (unsigned)(unsigned long long)p;
}
__device__ __forceinline__ void async_ld_b128(unsigned lds_byte, unsigned gofs,
                                              const float* base) {
  // GVS mode: copies 16B per lane from global base (SGPR) + offset gofs (VGPR) to the LDS address in lds_byte
  asm volatile("global_load_async_to_lds_b128 %0, %1, %2"
               :: "v"(lds_byte), "v"(gofs), "s"(base)
               : "memory");
}
__device__ __forceinline__ void async_wait0() {
  asm volatile("s_wait_asynccnt 0x0" ::: "memory");
}

// ---------------------------------------------------------------------------
// Stage 1: h1[d][n][h] = tanh(x[n][d] * W1[d][h] + b1[d][h])
// ---------------------------------------------------------------------------
__global__ __launch_bounds__(256) void spinn_stage1(
    const float* __restrict__ x,   // [256][3]
    const float* __restrict__ W1,  // [3][256]
    const float* __restrict__ b1,  // [3][256]
    float* __restrict__ h1) {      // [3][256][256]
  int tid = blockIdx.x * 256 + threadIdx.x;   // 0 .. 196607
  int d   = tid >> 16;
  int rem = tid & 0xFFFF;
  int n   = rem >> 8;
  int h   = rem & 0xFF;
  float v = x[n * 3 + d] * W1[d * 256 + h] + b1[d * 256 + h];
  h1[tid] = tanhf(v);
}

// ---------------------------------------------------------------------------
// Pre-transpose weights: OutT[d][k][h] = In[d][h][k]  (32x32 LDS-tiled)
// grid = (64, 3), block = 256
// ---------------------------------------------------------------------------
__global__ __launch_bounds__(256) void spinn_transpose(
    const float* __restrict__ In, float* __restrict__ OutT) {
  __shared__ float tile[32][33];
  int d  = blockIdx.y;
  int tx = blockIdx.x & 7, ty = blockIdx.x >> 3;
  int r0 = ty * 32, c0 = tx * 32;
  int lr = threadIdx.x >> 5, lc = threadIdx.x & 31;
  const float* src = In + d * 65536;
  float* dst = OutT + d * 65536;
  #pragma unroll
  for (int q = 0; q < 4; ++q)
    tile[lr + 8 * q][lc] = src[(r0 + lr + 8 * q) * 256 + c0 + lc];
  __syncthreads();
  #pragma unroll
  for (int q = 0; q < 4; ++q)
    dst[(c0 + lr + 8 * q) * 256 + r0 + lc] = tile[lc][lr + 8 * q];
}

// ---------------------------------------------------------------------------
// Stages 2/3: per-d GEMM  Out[d][n][k] = act( sum_h A[d][n][h]*WT[d][k][h] + bias[d][k] )
// W pre-transposed -> B fragments are contiguous float2 LDS reads.
// grid.x = 48 (d:3 x ntile:4 x ktile:4), block = 256 (8 waves).
// ---------------------------------------------------------------------------
__global__ __launch_bounds__(256) void spinn_gemm256(
    const float* __restrict__ Ain,   // [3][256][256]
    const float* __restrict__ WT,    // [3][256][256]  (k-major)
    const float* __restrict__ bias,  // [3][256]
    float* __restrict__ Out,         // [3][256][256]
    int doTanh) {
  __shared__ float ldsA[64 * 260];    // 64 n-rows, padded stride 260
  __shared__ float ldsWT[64 * 260];   // 64 k-rows, padded stride 260

  int bx = blockIdx.x;
  int d  = bx >> 4;
  int nt = (bx >> 2) & 3;
  int kt = bx & 3;
  int n0 = nt * 64, k0 = kt * 64;
  const float* Abase  = Ain + d * 65536 + n0 * 256;
  const float* WTbase = WT + d * 65536 + k0 * 256;
  int t = threadIdx.x;

  // Async fill: both 64x256 tiles, per-s advance (64B) identical in LDS/global
  {
    int r = t >> 2, c4 = t & 3;
    unsigned la = lds_off(ldsA)  + (unsigned)(r * 1040 + c4 * 16);
    unsigned lw = lds_off(ldsWT) + (unsigned)(r * 1040 + c4 * 16);
    unsigned go = (unsigned)(r * 1024 + c4 * 16);
    #pragma unroll
    for (int s = 0; s < 16; ++s) {
      async_ld_b128(la + s * 64, go + s * 64, Abase);
      async_ld_b128(lw + s * 64, go + s * 64, WTbase);
    }
  }
  async_wait0();
  __syncthreads();

  int wave = t >> 5;
  int lane = t & 31;
  int g = lane >> 4, i = lane & 15;   // f32 16x16x4 frag: lane holds K=2g,2g+1
  int rowsub  = wave & 3;
  int colhalf = wave >> 2;

  v8f acc0 = {};
  v8f acc1 = {};
  const float* arow = &ldsA[(rowsub * 16 + i) * 260];
  const float* wt0  = &ldsWT[(colhalf * 32 + i) * 260];
  const float* wt1  = &ldsWT[(colhalf * 32 + 16 + i) * 260];
  #pragma unroll 4
  for (int kk = 0; kk < 256; kk += 4) {
    int o = kk + 2 * g;
    v2f af = *(const v2f*)&arow[o];
    v2f b0 = *(const v2f*)&wt0[o];
    acc0 = __builtin_amdgcn_wmma_f32_16x16x4_f32(false, af, false, b0,
                                                 (short)0, acc0, false, false);
    v2f b1 = *(const v2f*)&wt1[o];
    acc1 = __builtin_amdgcn_wmma_f32_16x16x4_f32(false, af, false, b1,
                                                 (short)0, acc1, false, false);
  }

  float* obase = Out + d * 65536 + n0 * 256 + k0;
  #pragma unroll
  for (int j = 0; j < 2; ++j) {
    v8f acc = j ? acc1 : acc0;
    int col = colhalf * 32 + j * 16 + i;
    float bv = bias[d * 256 + k0 + col];
    #pragma unroll
    for (int p = 0; p < 8; ++p) {
      int row = rowsub * 16 + p + 8 * g;    // C/D layout: M = p + 8*(lane>=16)
      float v = acc[p] + bv;
      if (doTanh) v = tanhf(v);
      obase[row * 256 + col] = v;
    }
  }
}

// ---------------------------------------------------------------------------
// Stage 4: CP reconstruction.
// out[a][b][c][m] = sum_z A[a][m][z]*B[b][m][z]*C[c][m][z]
// WG (128 thr / 4 waves) = one a, 64-b tile, 64-c tile, both m.
// Wave = 2 row-subtiles x 4 col-subtiles of one m: 8 accumulators,
// 7 float2 LDS loads per K-step feed 8 WMMAs (C frags reused across rows).
// grid = (256, 16)
// ---------------------------------------------------------------------------
__global__ __launch_bounds__(128) void spinn_cp(
    const float* __restrict__ res,   // [3][256][256]  (d, n, m*128+z)
    float* __restrict__ out) {       // [256][256][256][2]
  __shared__ float ldsArow[256];
  __shared__ float ldsB[64 * 260];
  __shared__ float ldsC[64 * 260];

  int a  = blockIdx.x;
  int bt = blockIdx.y >> 2, ct = blockIdx.y & 3;
  int b0 = bt * 64, c0 = ct * 64;
  const float* Arow  = res + a * 256;
  const float* Bbase = res + 65536 + b0 * 256;
  const float* Cbase = res + 131072 + c0 * 256;
  int t = threadIdx.x;

  if (t < 64)
    async_ld_b128(lds_off(ldsArow) + (unsigned)(t * 16), (unsigned)(t * 16), Arow);
  {
    int r = t >> 1, c2 = t & 1;                       // 2 threads per 64 rows
    unsigned lb = lds_off(ldsB) + (unsigned)(r * 1040 + c2 * 16);
    unsigned lc = lds_off(ldsC) + (unsigned)(r * 1040 + c2 * 16);
    unsigned go = (unsigned)(r * 1024 + c2 * 16);
    #pragma unroll
    for (int s = 0; s < 32; ++s) {
      async_ld_b128(lb + s * 32, go + s * 32, Bbase);
      async_ld_b128(lc + s * 32, go + s * 32, Cbase);
    }
  }
  async_wait0();
  __syncthreads();

  int wave = t >> 5;
  int lane = t & 31;
  int g = lane >> 4, i = lane & 15;
  int m  = wave >> 1;       // waves 0-1 -> m=0, waves 2-3 -> m=1
  int rp = wave & 1;        // 32-row half of the 64 b-rows

  v8f acc[2][4] = {};
  const float* aptr = &ldsArow[m * 128];
  const float* bp0  = &ldsB[(rp * 32 + i) * 260 + m * 128];
  const float* bp1  = &ldsB[(rp * 32 + 16 + i) * 260 + m * 128];
  const float* cpt0 = &ldsC[(0 * 16 + i) * 260 + m * 128];
  const float* cpt1 = &ldsC[(1 * 16 + i) * 260 + m * 128];
  const float* cpt2 = &ldsC[(2 * 16 + i) * 260 + m * 128];
  const float* cpt3 = &ldsC[(3 * 16 + i) * 260 + m * 128];

  #pragma unroll 2
  for (int kk = 0; kk < 128; kk += 4) {
    int o = kk + 2 * g;
    v2f av  = *(const v2f*)&aptr[o];     // broadcast A[a][m][z..z+1]
    v2f bv0 = *(const v2f*)&bp0[o];
    v2f bv1 = *(const v2f*)&bp1[o];
    v2f af0 = av * bv0;                  // AB row fragments (2 muls each)
    v2f af1 = av * bv1;
    v2f cv;
    cv = *(const v2f*)&cpt0[o];
    acc[0][0] = __builtin_amdgcn_wmma_f32_16x16x4_f32(false, af0, false, cv,
                                                      (short)0, acc[0][0], false, false);
    acc[1][0] = __builtin_amdgcn_wmma_f32_16x16x4_f32(false, af1, false, cv,
                                                      (short)0, acc[1][0], false, false);
    cv = *(const v2f*)&cpt1[o];
    acc[0][1] = __builtin_amdgcn_wmma_f32_16x16x4_f32(false, af0, false, cv,
                                                      (short)0, acc[0][1], false, false);
    acc[1][1] = __builtin_amdgcn_wmma_f32_16x16x4_f32(false, af1, false, cv,
                                                      (short)0, acc[1][1], false, false);
    cv = *(const v2f*)&cpt2[o];
    acc[0][2] = __builtin_amdgcn_wmma_f32_16x16x4_f32(false, af0, false, cv,
                                                      (short)0, acc[0][2], false, false);
    acc[1][2] = __builtin_amdgcn_wmma_f32_16x16x4_f32(false, af1, false, cv,
                                                      (short)0, acc[1][2], false, false);
    cv = *(const v2f*)&cpt3[o];
    acc[0][3] = __builtin_amdgcn_wmma_f32_16x16x4_f32(false, af0, false, cv,
                                                      (short)0, acc[0][3], false, false);
    acc[1][3] = __builtin_amdgcn_wmma_f32_16x16x4_f32(false, af1, false, cv,
                                                      (short)0, acc[1][3], false, false);
  }

  // out[((a*256 + b)*256 + c)*2 + m]
  size_t obase = (((size_t)a * 256 + b0) * 256 + c0) * 2 + m;
  #pragma unroll
  for (int rs = 0; rs < 2; ++rs) {
    #pragma unroll
    for (int j = 0; j < 4; ++j) {
      int col = j * 16 + i;
      #pragma unroll
      for (int p = 0; p < 8; ++p) {
        int row = rp * 32 + rs * 16 + p + 8 * g;
        out[obase + (size_t)row * 512 + (size_t)col * 2] = acc[rs][j][p];
      }
    }
  }
}

// ---------------------------------------------------------------------------
extern "C" void kernel_launch(void* const* d_in, const int* in_sizes, int n_in,
                              void* d_out, int out_size, void* d_ws, size_t ws_size,
                              hipStream_t stream) {
  (void)in_sizes; (void)n_in; (void)out_size; (void)ws_size;
  const float* x  = (const float*)d_in[0];  // [256][3]
  const float* W1 = (const float*)d_in[1];  // [3][256]
  const float* b1 = (const float*)d_in[2];  // [3][256]
  const float* W2 = (const float*)d_in[3];  // [3][256][256]
  const float* b2 = (const float*)d_in[4];  // [3][256]
  const float* W3 = (const float*)d_in[5];  // [3][256][256]
  const float* b3 = (const float*)d_in[6];  // [3][256]
  float* out = (float*)d_out;               // [256][256][256][2]

  float* h1  = (float*)d_ws;                // [3][256][256]
  float* h2  = h1 + 3 * 65536;              // [3][256][256]
  float* res = h2 + 3 * 65536;              // [3][256][256]
  float* WT2 = res + 3 * 65536;             // [3][256][256] transposed
  float* WT3 = WT2 + 3 * 65536;             // [3][256][256] transposed

  spinn_stage1<<<768, 256, 0, stream>>>(x, W1, b1, h1);
  spinn_transpose<<<dim3(64, 3), 256, 0, stream>>>(W2, WT2);
  spinn_transpose<<<dim3(64, 3), 256, 0, stream>>>(W3, WT3);
  spinn_gemm256<<<48, 256, 0, stream>>>(h1, WT2, b2, h2, 1);
  spinn_gemm256<<<48, 256, 0, stream>>>(h2, WT3, b3, res, 0);
  spinn_cp<<<dim3(256, 16), 128, 0, stream>>>(res, out);
}